// PMEModelWrapper_72103910966065
// MI455X (gfx1250) — compile-verified
//
#include <hip/hip_runtime.h>
#include <hip/hip_bf16.h>
#include <math.h>

#define KMESH 64
#define M3D (KMESH * KMESH * KMESH)
#define ALPHA_C 0.35f
#define KE_CONST 14.399645478425668f
#define PI_F 3.14159265358979323846f
#define TWO_OVER_SQRTPI 1.1283791670955126f

// ---- CDNA5 async global->LDS copy availability (probe via __has_builtin) ----
#if defined(__has_builtin)
#  if __has_builtin(__builtin_amdgcn_global_load_async_to_lds_b32)
#    define HAVE_ASYNC_LDS 1
#  endif
#  if __has_builtin(__builtin_amdgcn_s_wait_asynccnt)
#    define HAVE_WAIT_ASYNC 1
#  endif
#endif
#ifndef HAVE_ASYNC_LDS
#  define HAVE_ASYNC_LDS 0
#endif
#ifndef HAVE_WAIT_ASYNC
#  define HAVE_WAIT_ASYNC 0
#endif

typedef __attribute__((ext_vector_type(2))) float v2f;
typedef __attribute__((ext_vector_type(8))) float v8f;

// ---------------------------------------------------------------- helpers
__device__ __forceinline__ float inv3x3(const float* c, float inv[9]) {
    float a00 = c[0], a01 = c[1], a02 = c[2];
    float a10 = c[3], a11 = c[4], a12 = c[5];
    float a20 = c[6], a21 = c[7], a22 = c[8];
    float c00 =  (a11 * a22 - a12 * a21);
    float c01 = -(a10 * a22 - a12 * a20);
    float c02 =  (a10 * a21 - a11 * a20);
    float det = a00 * c00 + a01 * c01 + a02 * c02;
    float id = 1.0f / det;
    inv[0] =  (a11 * a22 - a12 * a21) * id;
    inv[1] = -(a01 * a22 - a02 * a21) * id;
    inv[2] =  (a01 * a12 - a02 * a11) * id;
    inv[3] = -(a10 * a22 - a12 * a20) * id;
    inv[4] =  (a00 * a22 - a02 * a20) * id;
    inv[5] = -(a00 * a12 - a02 * a10) * id;
    inv[6] =  (a10 * a21 - a11 * a20) * id;
    inv[7] = -(a00 * a21 - a01 * a20) * id;
    inv[8] =  (a00 * a11 - a01 * a10) * id;
    return det;
}

// cardinal B-splines, order 4, and derivative
__device__ __forceinline__ float M2f(float x) {
    float t = 1.0f - fabsf(x - 1.0f);
    return t > 0.0f ? t : 0.0f;
}
__device__ __forceinline__ float M3f(float x) {
    return (x * M2f(x) + (3.0f - x) * M2f(x - 1.0f)) * 0.5f;
}
__device__ __forceinline__ float M4f(float x) {
    return (x * M3f(x) + (4.0f - x) * M3f(x - 1.0f)) * (1.0f / 3.0f);
}
__device__ __forceinline__ float dM4f(float x) {
    return M3f(x) - M3f(x - 1.0f);
}

// |b(m)|^-2 Euler-spline factor for K=64, order 4 (Mv = 1/6, 2/3, 1/6)
__device__ __forceinline__ float bsp_fac(int m) {
    float base = 2.0f * PI_F * (float)m / 64.0f;
    float s1, c1, s2, c2;
    __sincosf(base, &s1, &c1);
    __sincosf(2.0f * base, &s2, &c2);
    float dre = (1.0f / 6.0f) + (2.0f / 3.0f) * c1 + (1.0f / 6.0f) * c2;
    float dim = (2.0f / 3.0f) * s1 + (1.0f / 6.0f) * s2;
    return 1.0f / (dre * dre + dim * dim);
}

// acc slots: 0 e_real, 1 e_rec, 2 qtot, 3 sumq2,
//            4..9 vir_real(xx,yy,zz,xy,xz,yz), 10..15 vir_rec

// ---------------------------------------------------------------- charge reduce
__global__ __launch_bounds__(256)
void reduceq_kernel(const float* __restrict__ q, float* __restrict__ acc, int N) {
    __shared__ float sred[256];
    int i = blockIdx.x * 256 + threadIdx.x;
    float qv = (i < N) ? q[i] : 0.0f;
    float vals[2] = {qv, qv * qv};
    const int slot[2] = {2, 3};
    for (int t = 0; t < 2; ++t) {
        sred[threadIdx.x] = vals[t];
        __syncthreads();
        for (int s = 128; s > 0; s >>= 1) {
            if (threadIdx.x < s) sred[threadIdx.x] += sred[threadIdx.x + s];
            __syncthreads();
        }
        if (threadIdx.x == 0) atomicAdd(&acc[slot[t]], sred[0]);
        __syncthreads();
    }
}

// ---------------------------------------------------------------- real-space pairs
__global__ __launch_bounds__(256)
void realspace_kernel(const float* __restrict__ pos, const float* __restrict__ q,
                      const float* __restrict__ cell, const int* __restrict__ nbr,
                      const int* __restrict__ shifts, float* __restrict__ grad,
                      float* __restrict__ acc, int N, int NNBR) {
    __shared__ float sred[256];
    int i = blockIdx.x * 256 + threadIdx.x;
    float e = 0.0f, gx = 0.0f, gy = 0.0f, gz = 0.0f;
    float vxx = 0.0f, vyy = 0.0f, vzz = 0.0f, vxy = 0.0f, vxz = 0.0f, vyz = 0.0f;
    if (i < N) {
        float pix = pos[3 * i], piy = pos[3 * i + 1], piz = pos[3 * i + 2];
        float qi = q[i];
        for (int jj = 0; jj < NNBR; ++jj) {
            size_t p = (size_t)i * NNBR + jj;
            int j = nbr[p];
            int sx = shifts[3 * p], sy = shifts[3 * p + 1], sz = shifts[3 * p + 2];
            bool self0 = (j == i) && (sx == 0) && (sy == 0) && (sz == 0);
            if (j < 0 || self0) continue;
            float rx = pos[3 * j]     - pix + sx * cell[0] + sy * cell[3] + sz * cell[6];
            float ry = pos[3 * j + 1] - piy + sx * cell[1] + sy * cell[4] + sz * cell[7];
            float rz = pos[3 * j + 2] - piz + sx * cell[2] + sy * cell[5] + sz * cell[8];
            float r2 = rx * rx + ry * ry + rz * rz;
            float r = sqrtf(r2);
            float inv_r = 1.0f / r;
            float qq = qi * q[j];
            float ec = erfcf(ALPHA_C * r);
            e += 0.5f * qq * ec * inv_r;
            // d/dr [erfc(ar)/r]
            float dphi = -(ec * inv_r * inv_r)
                         - TWO_OVER_SQRTPI * ALPHA_C * expf(-ALPHA_C * ALPHA_C * r2) * inv_r;
            float g = 0.5f * qq * dphi * inv_r;   // dE/dr_j = +g*rvec ; dE/dr_i = -g*rvec
            gx -= g * rx; gy -= g * ry; gz -= g * rz;
            atomicAdd(&grad[3 * j + 0], g * rx);
            atomicAdd(&grad[3 * j + 1], g * ry);
            atomicAdd(&grad[3 * j + 2], g * rz);
            vxx += g * rx * rx; vyy += g * ry * ry; vzz += g * rz * rz;
            vxy += g * rx * ry; vxz += g * rx * rz; vyz += g * ry * rz;
        }
        atomicAdd(&grad[3 * i + 0], gx);
        atomicAdd(&grad[3 * i + 1], gy);
        atomicAdd(&grad[3 * i + 2], gz);
    }
    float vals[7] = {e, vxx, vyy, vzz, vxy, vxz, vyz};
    const int slot[7] = {0, 4, 5, 6, 7, 8, 9};
    for (int t = 0; t < 7; ++t) {
        sred[threadIdx.x] = vals[t];
        __syncthreads();
        for (int s = 128; s > 0; s >>= 1) {
            if (threadIdx.x < s) sred[threadIdx.x] += sred[threadIdx.x + s];
            __syncthreads();
        }
        if (threadIdx.x == 0) atomicAdd(&acc[slot[t]], sred[0]);
        __syncthreads();
    }
}

// ---------------------------------------------------------------- charge spreading
__global__ __launch_bounds__(256)
void spread_kernel(const float* __restrict__ pos, const float* __restrict__ q,
                   const float* __restrict__ cell, float* __restrict__ mesh, int N) {
    int i = blockIdx.x * 256 + threadIdx.x;
    if (i >= N) return;
    float invc[9];
    inv3x3(cell, invc);
    float px = pos[3 * i], py = pos[3 * i + 1], pz = pos[3 * i + 2];
    float qi = q[i];
    float w[3][4];
    int gi[3][4];
    for (int d = 0; d < 3; ++d) {
        float fr = px * invc[0 * 3 + d] + py * invc[1 * 3 + d] + pz * invc[2 * 3 + d];
        float u = fr * 64.0f;
        float fl = floorf(u);
        float t = u - fl;
        int b = (int)fl;
        for (int o = 0; o < 4; ++o) {
            w[d][o] = M4f(t + (float)o);
            int m = (b - o) % 64;
            if (m < 0) m += 64;
            gi[d][o] = m;
        }
    }
    for (int ox = 0; ox < 4; ++ox)
        for (int oy = 0; oy < 4; ++oy) {
            float wxy = qi * w[0][ox] * w[1][oy];
            int baseix = (gi[0][ox] * 64 + gi[1][oy]) * 64;
            for (int oz = 0; oz < 4; ++oz)
                atomicAdd(&mesh[baseix + gi[2][oz]], wxy * w[2][oz]);
        }
}

// ---------------------------------------------------------------- DFT pass via WMMA
// Out(4096 lines x 64) = In x W, W[n][k] = exp(sgn*2*pi*i*n*k/64) (W symmetric).
// One wave32 per 16-line group: stage the 16x64 A-tile into LDS once (async copy
// on CDNA5, ASYNCcnt), then compute all four 16x16 k-tiles out of LDS (4x reuse).
// MODE: 0 = real input forward (Ai==0), 1 = full complex, 2 = real-output inverse.
template <int MODE>
__global__ __launch_bounds__(32)
void dft_pass_kernel(const float* __restrict__ inR, const float* __restrict__ inI,
                     float* __restrict__ outR, float* __restrict__ outI,
                     int inner, float sgn) {
    __shared__ float sAR[1024];
    __shared__ float sAI[1024];
    const int lineBase = blockIdx.x * 16;   // 256 line groups
    const int lane = threadIdx.x;
    const int lo = lane & 15;
    const int hi = lane >> 4;

    // ---- stage A tile (16 rows x 64 elems) into LDS ----
    for (int it = 0; it < 32; ++it) {
        int e = lane + 32 * it;             // 0..1023
        int r = lineBase + (e >> 6);
        int n = e & 63;
        int ga = (r / inner) * 64 * inner + (r % inner) + n * inner;
#if HAVE_ASYNC_LDS
        __builtin_amdgcn_global_load_async_to_lds_b32(
            (__attribute__((address_space(1))) int*)(inR + ga),
            (__attribute__((address_space(3))) int*)&sAR[e], 0, 0);
        if (MODE != 0)
            __builtin_amdgcn_global_load_async_to_lds_b32(
                (__attribute__((address_space(1))) int*)(inI + ga),
                (__attribute__((address_space(3))) int*)&sAI[e], 0, 0);
#else
        sAR[e] = inR[ga];
        if (MODE != 0) sAI[e] = inI[ga];
#endif
    }
#if HAVE_ASYNC_LDS
#  if HAVE_WAIT_ASYNC
    __builtin_amdgcn_s_wait_asynccnt(0);
#  else
    asm volatile("s_wait_asynccnt 0x0" ::: "memory");
#  endif
#endif
    __syncthreads();

    const int rowLDS = lo << 6;             // A row for this lane (M = lane%16)
    const float wstep = sgn * 2.0f * PI_F / 64.0f;

    for (int kt = 0; kt < 4; ++kt) {
        const int kBase = kt * 16;
        const int colB = kBase + lo;        // B column (N = lane%16)
        v8f cr = {};
        v8f ci = {};
        for (int kk = 0; kk < 16; ++kk) {
            int n0 = kk * 4 + 2 * hi;       // K = v + 2*hi per ISA A/B layout
            v2f ar = *reinterpret_cast<const v2f*>(&sAR[rowLDS + n0]);
            v2f ai = {};
            if (MODE != 0) ai = *reinterpret_cast<const v2f*>(&sAI[rowLDS + n0]);

            float s0, c0, s1, c1;
            __sincosf(wstep * (float)(n0 * colB), &s0, &c0);
            __sincosf(wstep * (float)((n0 + 1) * colB), &s1, &c1);
            v2f br; br.x = c0; br.y = c1;
            v2f bi; bi.x = s0; bi.y = s1;

            // Cr += Ar*Br - Ai*Bi ; Ci += Ar*Bi + Ai*Br
            cr = __builtin_amdgcn_wmma_f32_16x16x4_f32(false, ar, false, br, (short)0, cr, false, false);
            if (MODE != 0) {
                v2f bin; bin.x = -s0; bin.y = -s1;
                cr = __builtin_amdgcn_wmma_f32_16x16x4_f32(false, ai, false, bin, (short)0, cr, false, false);
            }
            if (MODE != 2) {
                ci = __builtin_amdgcn_wmma_f32_16x16x4_f32(false, ar, false, bi, (short)0, ci, false, false);
                if (MODE != 0)
                    ci = __builtin_amdgcn_wmma_f32_16x16x4_f32(false, ai, false, br, (short)0, ci, false, false);
            }
        }
        // D layout: VGPR v -> M = v + 8*hi, N = lane%16
        for (int v = 0; v < 8; ++v) {
            int row = lineBase + v + 8 * hi;
            int k = kBase + lo;
            int o = (row / inner) * 64 * inner + (row % inner) + k * inner;
            outR[o] = cr[v];
            if (MODE != 2) outI[o] = ci[v];
        }
    }
}

// ---------------------------------------------------------------- k-space multiply
__global__ __launch_bounds__(256)
void kspace_kernel(float* __restrict__ Qr, float* __restrict__ Qi,
                   const float* __restrict__ cell, float* __restrict__ acc) {
    __shared__ float sred[256];
    int idx = blockIdx.x * 256 + threadIdx.x;      // < 64^3, exact
    int mz = idx & 63, my = (idx >> 6) & 63, mx = idx >> 12;
    float invc[9];
    float det = inv3x3(cell, invc);
    float vol = fabsf(det);
    float fx = (mx < 32) ? (float)mx : (float)(mx - 64);
    float fy = (my < 32) ? (float)my : (float)(my - 64);
    float fz = (mz < 32) ? (float)mz : (float)(mz - 64);
    float twopi = 2.0f * PI_F;
    float kx = twopi * (fx * invc[0] + fy * invc[1] + fz * invc[2]);
    float ky = twopi * (fx * invc[3] + fy * invc[4] + fz * invc[5]);
    float kz = twopi * (fx * invc[6] + fy * invc[7] + fz * invc[8]);
    float k2 = kx * kx + ky * ky + kz * kz;

    float qr = Qr[idx], qi = Qi[idx];
    float epart = 0.0f, vxx = 0.0f, vyy = 0.0f, vzz = 0.0f, vxy = 0.0f, vxz = 0.0f, vyz = 0.0f;
    float a2 = ALPHA_C * ALPHA_C;
    if (k2 > 0.0f) {
        float kern = expf(-k2 / (4.0f * a2)) / k2;
        float Bk = kern * bsp_fac(mx) * bsp_fac(my) * bsp_fac(mz);
        float pref = twopi / vol;
        float q2 = qr * qr + qi * qi;
        epart = pref * Bk * q2;
        float fac = 2.0f * (1.0f / (4.0f * a2) + 1.0f / k2);
        vxx = epart * (fac * kx * kx - 1.0f);
        vyy = epart * (fac * ky * ky - 1.0f);
        vzz = epart * (fac * kz * kz - 1.0f);
        vxy = epart * (fac * kx * ky);
        vxz = epart * (fac * kx * kz);
        vyz = epart * (fac * ky * kz);
        float scale = 2.0f * pref * Bk;   // so unnormalized inverse DFT gives dE/dq_m
        Qr[idx] = qr * scale;
        Qi[idx] = qi * scale;
    } else {
        Qr[idx] = 0.0f;
        Qi[idx] = 0.0f;
    }

    float vals[7] = {epart, vxx, vyy, vzz, vxy, vxz, vyz};
    const int slot[7] = {1, 10, 11, 12, 13, 14, 15};
    for (int t = 0; t < 7; ++t) {
        sred[threadIdx.x] = vals[t];
        __syncthreads();
        for (int s = 128; s > 0; s >>= 1) {
            if (threadIdx.x < s) sred[threadIdx.x] += sred[threadIdx.x + s];
            __syncthreads();
        }
        if (threadIdx.x == 0) atomicAdd(&acc[slot[t]], sred[0]);
        __syncthreads();
    }
}

// ---------------------------------------------------------------- force gather
__global__ __launch_bounds__(256)
void gather_kernel(const float* __restrict__ pos, const float* __restrict__ q,
                   const float* __restrict__ cell, const float* __restrict__ phi,
                   float* __restrict__ grad, int N) {
    int i = blockIdx.x * 256 + threadIdx.x;
    if (i >= N) return;
    float invc[9];
    inv3x3(cell, invc);
    float px = pos[3 * i], py = pos[3 * i + 1], pz = pos[3 * i + 2];
    float qi = q[i];
    float w[3][4], dw[3][4];
    int gi[3][4];
    for (int d = 0; d < 3; ++d) {
        float fr = px * invc[0 * 3 + d] + py * invc[1 * 3 + d] + pz * invc[2 * 3 + d];
        float u = fr * 64.0f;
        float fl = floorf(u);
        float t = u - fl;
        int b = (int)fl;
        for (int o = 0; o < 4; ++o) {
            w[d][o] = M4f(t + (float)o);
            dw[d][o] = dM4f(t + (float)o);
            int m = (b - o) % 64;
            if (m < 0) m += 64;
            gi[d][o] = m;
        }
    }
    float gu0 = 0.0f, gu1 = 0.0f, gu2 = 0.0f;
    for (int ox = 0; ox < 4; ++ox)
        for (int oy = 0; oy < 4; ++oy) {
            int baseix = (gi[0][ox] * 64 + gi[1][oy]) * 64;
            for (int oz = 0; oz < 4; ++oz) {
                float p = phi[baseix + gi[2][oz]];
                gu0 += p * dw[0][ox] * w[1][oy] * w[2][oz];
                gu1 += p * w[0][ox] * dw[1][oy] * w[2][oz];
                gu2 += p * w[0][ox] * w[1][oy] * dw[2][oz];
            }
        }
    gu0 *= qi * 64.0f; gu1 *= qi * 64.0f; gu2 *= qi * 64.0f;
    for (int a = 0; a < 3; ++a)
        grad[3 * i + a] += gu0 * invc[a * 3 + 0] + gu1 * invc[a * 3 + 1] + gu2 * invc[a * 3 + 2];
}

// ---------------------------------------------------------------- finalize
__global__ __launch_bounds__(256)
void finalize_forces_kernel(const float* __restrict__ grad, float* __restrict__ out, int n3) {
    int t = blockIdx.x * 256 + threadIdx.x;
    if (t < n3) out[1 + t] = -KE_CONST * grad[t];
}

__global__ void finalize_scalar_kernel(const float* __restrict__ acc,
                                       const float* __restrict__ cell,
                                       float* __restrict__ out, int N) {
    float invc[9];
    float det = inv3x3(cell, invc);
    float vol = fabsf(det);
    float e_real = acc[0], e_rec = acc[1], qtot = acc[2], sumq2 = acc[3];
    float a2 = ALPHA_C * ALPHA_C;
    float e_self = -(ALPHA_C / sqrtf(PI_F)) * sumq2;
    float e_bg = -(PI_F / (2.0f * a2 * vol)) * qtot * qtot;
    out[0] = (e_real + e_rec + e_self + e_bg) * KE_CONST;
    // dE/d(eps) (symmetric): real + reciprocal + background (-e_bg * I)
    float g[9];
    g[0] = acc[4] + acc[10] - e_bg;
    g[4] = acc[5] + acc[11] - e_bg;
    g[8] = acc[6] + acc[12] - e_bg;
    g[1] = g[3] = acc[7] + acc[13];
    g[2] = g[6] = acc[8] + acc[14];
    g[5] = g[7] = acc[9] + acc[15];
    int base = 1 + 3 * N;
    for (int t = 0; t < 9; ++t) out[base + t] = -g[t] * KE_CONST / vol;
}

// ---------------------------------------------------------------- launch
extern "C" void kernel_launch(void* const* d_in, const int* in_sizes, int n_in,
                              void* d_out, int out_size, void* d_ws, size_t ws_size,
                              hipStream_t stream) {
    const float* pos  = (const float*)d_in[0];
    const float* q    = (const float*)d_in[1];
    const float* cell = (const float*)d_in[2];
    const int* nbr    = (const int*)d_in[3];
    const int* shf    = (const int*)d_in[4];
    int N = in_sizes[0] / 3;
    int NNBR = (N > 0) ? in_sizes[3] / N : 0;

    float* ws = (float*)d_ws;
    float* acc = ws;                                        // 16 floats
    float* grad = ws + 16;                                  // 3N floats
    size_t meshOff = ((size_t)(16 + 3 * N) + 63) & ~(size_t)63;
    float* mesh = ws + meshOff;                             // 64^3
    float* Ar = mesh + M3D;
    float* Ai = Ar + M3D;
    float* Br = Ai + M3D;
    float* Bi = Br + M3D;

    // zero accumulators + gradient + mesh (complex buffers fully overwritten)
    (void)hipMemsetAsync(d_ws, 0, (meshOff + (size_t)M3D) * sizeof(float), stream);

    int nb = (N + 255) / 256;
    reduceq_kernel<<<nb, 256, 0, stream>>>(q, acc, N);
    realspace_kernel<<<nb, 256, 0, stream>>>(pos, q, cell, nbr, shf, grad, acc, N, NNBR);
    spread_kernel<<<nb, 256, 0, stream>>>(pos, q, cell, mesh, N);

    // forward FFT (DFT-by-WMMA) along z, y, x
    dft_pass_kernel<0><<<256, 32, 0, stream>>>(mesh, nullptr, Ar, Ai, 1, -1.0f);
    dft_pass_kernel<1><<<256, 32, 0, stream>>>(Ar, Ai, Br, Bi, 64, -1.0f);
    dft_pass_kernel<1><<<256, 32, 0, stream>>>(Br, Bi, Ar, Ai, 4096, -1.0f);

    kspace_kernel<<<M3D / 256, 256, 0, stream>>>(Ar, Ai, cell, acc);

    // inverse (unnormalized) FFT -> dE/dq mesh in Br (real part)
    dft_pass_kernel<1><<<256, 32, 0, stream>>>(Ar, Ai, Br, Bi, 1, 1.0f);
    dft_pass_kernel<1><<<256, 32, 0, stream>>>(Br, Bi, Ar, Ai, 64, 1.0f);
    dft_pass_kernel<2><<<256, 32, 0, stream>>>(Ar, Ai, Br, Bi, 4096, 1.0f);

    gather_kernel<<<nb, 256, 0, stream>>>(pos, q, cell, Br, grad, N);

    finalize_forces_kernel<<<(3 * N + 255) / 256, 256, 0, stream>>>(grad, (float*)d_out, 3 * N);
    finalize_scalar_kernel<<<1, 1, 0, stream>>>(acc, cell, (float*)d_out, N);
}